// HopInvariantLayer_618475291107
// MI455X (gfx1250) — compile-verified
//
#include <hip/hip_runtime.h>

// CDNA5 / gfx1250: wave32, WMMA F32 16x16x4 (A,B = 2 VGPRs f32; C/D = 8 VGPRs f32)
typedef __attribute__((ext_vector_type(2))) float v2f;
typedef __attribute__((ext_vector_type(8))) float v8f;

#define BLK  128     // threads per block = points per block (4 waves)
#define FSTR 20      // LDS row stride (floats) for staged features: 16 + pad, 16B aligned
#define PSTR 68      // LDS row stride (floats) for projection rows: 64 + pad (bank decorrelation)

__global__ __launch_bounds__(BLK)
void hop_invariant_kernel(const float* __restrict__ tf,
                          const float* __restrict__ C2,   // (3,3,5)  flat: [(i*3+j)*5+m]
                          const float* __restrict__ C3,   // (3,3,3,7) flat: [((i*3+j)*3+k)*7+m]
                          float* __restrict__ out,        // (N,13)
                          int n)
{
    __shared__ float Wlds[16 * 64];        //  4 KB : fused projection matrix (16 x 51, padded)
    __shared__ float Flds[BLK * FSTR];     // 10 KB : per-point features
    __shared__ float Plds[BLK * PSTR];     // 34 KB : per-point projections (51 used cols)

    const int tid = threadIdx.x;

    // ---- Phase 0: build W in LDS. cols 0..8 -> a_half, 9..35 -> b_half, 36..50 -> e_half.
    if (tid < 64) {
        const int c = tid;
        #pragma unroll
        for (int r = 0; r < 16; ++r) Wlds[r * 64 + c] = 0.0f;
        if (c < 9) {                       // a_half[i][j] = sum_m q[m]*C2[i,j,m]
            #pragma unroll
            for (int m = 0; m < 5; ++m) Wlds[(4 + m) * 64 + c] = C2[c * 5 + m];
        } else if (c < 36) {               // b_half[i][j][k] = sum_m t[m]*C3[i,j,k,m]
            const int ijk = c - 9;
            #pragma unroll
            for (int m = 0; m < 7; ++m) Wlds[(9 + m) * 64 + c] = C3[ijk * 7 + m];
        } else if (c < 51) {               // e_half[k][n] = sum_m t[m]*sum_ij C3[i,j,k,m]*C2[i,j,n]
            const int kn = c - 36;
            const int k = kn / 5, n5 = kn % 5;
            #pragma unroll
            for (int m = 0; m < 7; ++m) {
                float acc = 0.0f;
                #pragma unroll
                for (int ij = 0; ij < 9; ++ij)
                    acc += C3[(ij * 3 + k) * 7 + m] * C2[ij * 5 + n5];
                Wlds[(9 + m) * 64 + c] = acc;
            }
        }
    }

    // ---- Phase 1: coalesced global -> LDS stage of this block's 128 points.
    const long pt = (long)blockIdx.x * BLK + tid;
    {
        float4 f0 = {}, f1 = {}, f2 = {}, f3 = {};
        if (pt < n) {
            const float4* src = (const float4*)(tf + pt * 16);
            f0 = src[0]; f1 = src[1]; f2 = src[2]; f3 = src[3];
        }
        float4* dst = (float4*)(&Flds[tid * FSTR]);   // 80B row stride: 16B aligned
        dst[0] = f0; dst[1] = f1; dst[2] = f2; dst[3] = f3;
    }
    __syncthreads();

    // ---- Phase 2: per-wave WMMA. proj(16pts x 64) = F(16 x 16) @ W(16 x 64).
    {
        const int wave = tid >> 5;
        const int lane = tid & 31;
        const int hi   = lane >> 4;   // selects K pair within a K=4 step (ISA 16x4 f32 layout)
        const int nn   = lane & 15;   // A row (M) / B,D column (N)

        // B fragments: W[K][col], shared by both point tiles of this wave.
        v2f Bfrag[4][4];
        #pragma unroll
        for (int t4 = 0; t4 < 4; ++t4) {
            #pragma unroll
            for (int kk = 0; kk < 4; ++kk) {
                const int col = t4 * 16 + nn;
                v2f b;
                b.x = Wlds[(kk * 4 + 2 * hi    ) * 64 + col];
                b.y = Wlds[(kk * 4 + 2 * hi + 1) * 64 + col];
                Bfrag[t4][kk] = b;
            }
        }

        #pragma unroll
        for (int tile = 0; tile < 2; ++tile) {
            const int pbase = wave * 32 + tile * 16;   // 16-point tile base (local)
            v2f Afrag[4];
            #pragma unroll
            for (int kk = 0; kk < 4; ++kk) {
                v2f a;
                a.x = Flds[(pbase + nn) * FSTR + kk * 4 + 2 * hi];
                a.y = Flds[(pbase + nn) * FSTR + kk * 4 + 2 * hi + 1];
                Afrag[kk] = a;
            }
            #pragma unroll
            for (int t4 = 0; t4 < 4; ++t4) {
                v8f acc = (v8f)0.0f;
                #pragma unroll
                for (int kk = 0; kk < 4; ++kk)
                    acc = __builtin_amdgcn_wmma_f32_16x16x4_f32(
                        false, Afrag[kk], false, Bfrag[t4][kk],
                        (short)0, acc, false, false);
                // D layout: VGPR r, lanes 0-15 -> (M=r, N=lane); lanes 16-31 -> (M=r+8, N=lane-16)
                #pragma unroll
                for (int r = 0; r < 8; ++r)
                    Plds[(pbase + r + 8 * hi) * PSTR + t4 * 16 + nn] = acc[r];
            }
        }
    }
    __syncthreads();

    // ---- Phase 3: per-point invariants (scalar VALU, ~400 flops/pt).
    if (pt < n) {
        const float* Fr = &Flds[tid * FSTR];
        const float* Pr = &Plds[tid * PSTR];

        const float s0 = Fr[0];
        float vv[3];
        #pragma unroll
        for (int i = 0; i < 3; ++i) vv[i] = Fr[1 + i];
        float qq[5];
        #pragma unroll
        for (int m = 0; m < 5; ++m) qq[m] = Fr[4 + m];

        float ah[9];                                       // ah[i*3+j]
        #pragma unroll
        for (int c = 0; c < 9;  ++c) ah[c] = Pr[c];
        float bh[27];                                      // bh[(i*3+j)*3+k]
        #pragma unroll
        for (int c = 0; c < 27; ++c) bh[c] = Pr[9 + c];
        float eh[15];                                      // eh[k*5+n]
        #pragma unroll
        for (int c = 0; c < 15; ++c) eh[c] = Pr[36 + c];

        const float I0 = s0;
        const float I1 = vv[0]*vv[0] + vv[1]*vv[1] + vv[2]*vv[2];

        // a[j][k] = sum_i ah[i][j]*ah[i][k]
        float am[9];
        #pragma unroll
        for (int j = 0; j < 3; ++j) {
            #pragma unroll
            for (int k = 0; k < 3; ++k)
                am[j*3+k] = ah[0*3+j]*ah[0*3+k] + ah[1*3+j]*ah[1*3+k] + ah[2*3+j]*ah[2*3+k];
        }
        const float I2 = am[0] + am[4] + am[8];

        // I3 = sum_m q[m] * sum_jk a[jk]*C2[j,k,m]   (C2 read back from W rows 4..8, broadcast)
        float I3 = 0.0f;
        #pragma unroll
        for (int m = 0; m < 5; ++m) {
            float acc = 0.0f;
            #pragma unroll
            for (int jk = 0; jk < 9; ++jk) acc += am[jk] * Wlds[(4 + m) * 64 + jk];
            I3 += acc * qq[m];
        }

        float cv[3];
        #pragma unroll
        for (int j = 0; j < 3; ++j)
            cv[j] = ah[j]*vv[0] + ah[3+j]*vv[1] + ah[6+j]*vv[2];
        const float I6 = cv[0]*vv[0] + cv[1]*vv[1] + cv[2]*vv[2];
        const float I7 = cv[0]*cv[0] + cv[1]*cv[1] + cv[2]*cv[2];

        // b[k][l] = sum_ij bh[ij][k]*bh[ij][l]
        float bm[9];
        #pragma unroll
        for (int k = 0; k < 3; ++k) {
            #pragma unroll
            for (int l = 0; l < 3; ++l) {
                float acc = 0.0f;
                #pragma unroll
                for (int ij = 0; ij < 9; ++ij) acc += bh[ij*3+k] * bh[ij*3+l];
                bm[k*3+l] = acc;
            }
        }
        const float I4 = bm[0] + bm[4] + bm[8];

        float I5 = 0.0f, I11 = 0.0f;
        #pragma unroll
        for (int c = 0; c < 9; ++c) { I5 += bm[c]*bm[c]; I11 += bm[c]*am[c]; }

        float I10 = 0.0f;
        #pragma unroll
        for (int m = 0; m < 5; ++m) {
            float acc = 0.0f;
            #pragma unroll
            for (int kl = 0; kl < 9; ++kl) acc += bm[kl] * Wlds[(4 + m) * 64 + kl];
            I10 += acc * qq[m];
        }

        float I8 = 0.0f, I9 = 0.0f;
        #pragma unroll
        for (int i = 0; i < 3; ++i) {
            #pragma unroll
            for (int j = 0; j < 3; ++j) {
                const float dij = bh[(i*3+j)*3+0]*vv[0] + bh[(i*3+j)*3+1]*vv[1] + bh[(i*3+j)*3+2]*vv[2];
                const float vij = vv[i] * vv[j];
                I8 += vij * dij;
                I9 += vij * bm[i*3+j];
            }
        }

        float e[3];
        #pragma unroll
        for (int k = 0; k < 3; ++k) {
            float acc = 0.0f;
            #pragma unroll
            for (int n5 = 0; n5 < 5; ++n5) acc += eh[k*5+n5] * qq[n5];
            e[k] = acc;
        }
        const float I12 = e[0]*e[0] + e[1]*e[1] + e[2]*e[2];

        float* o = out + pt * 13;
        o[0]=I0;  o[1]=I1;  o[2]=I2;  o[3]=I3;  o[4]=I4;  o[5]=I5;  o[6]=I6;
        o[7]=I7;  o[8]=I8;  o[9]=I9;  o[10]=I10; o[11]=I11; o[12]=I12;
    }
}

extern "C" void kernel_launch(void* const* d_in, const int* in_sizes, int n_in,
                              void* d_out, int out_size, void* d_ws, size_t ws_size,
                              hipStream_t stream) {
    (void)n_in; (void)out_size; (void)d_ws; (void)ws_size;
    const float* tf = (const float*)d_in[0];
    const float* C2 = (const float*)d_in[1];
    const float* C3 = (const float*)d_in[2];
    float* out = (float*)d_out;
    const int n = in_sizes[0] / 16;          // N points (2,000,000)
    const int blocks = (n + BLK - 1) / BLK;  // 15625, exact multiple
    hipLaunchKernelGGL(hop_invariant_kernel, dim3(blocks), dim3(BLK), 0, stream,
                       tf, C2, C3, out, n);
}